// shared_model_attention_43044162240998
// MI455X (gfx1250) — compile-verified
//
#include <hip/hip_runtime.h>
#include <hip/hip_bf16.h>
#include <math.h>

typedef __attribute__((ext_vector_type(2))) float v2f;
typedef __attribute__((ext_vector_type(8))) float v8f;

#define B_N 2048
#define E_N 102400
#define F_N 512
#define H_N 100
#define H_PAD 112
#define NTHREADS 224
#define KSPLIT 16

// Issue one 16-byte async global->LDS copy (ASYNCcnt-tracked, no VGPR round
// trip). ldsoff = byte offset in LDS (low 32 bits of flat shared address).
__device__ __forceinline__ void async_copy_b128(unsigned ldsoff, const float* g) {
    asm volatile("global_load_async_to_lds_b128 %0, %1, off"
                 :: "v"(ldsoff), "v"(g) : "memory");
}
__device__ __forceinline__ void async_wait0() {
    asm volatile("s_wait_asynccnt 0x0" ::: "memory");
}

// ---------------------------------------------------------------------------
__global__ void zero_kernel(float* __restrict__ p, int n) {
    int i = blockIdx.x * 256 + threadIdx.x;
    if (i < n) p[i] = 0.0f;
}

// ---------------------------------------------------------------------------
// 3-layer tanh MLP: 16 rows per block of X[.,512] -> out_pad[.,112].
// Branch-free inner loops (clamped address * 0/1 mask); layers 2/3 truncate
// K to 100 because padded activation columns are exactly zero.
// ---------------------------------------------------------------------------
__global__ __launch_bounds__(NTHREADS) void mlp_kernel(
    const float* __restrict__ X,
    const float* __restrict__ W1, const float* __restrict__ b1,
    const float* __restrict__ W2, const float* __restrict__ b2,
    const float* __restrict__ W3, const float* __restrict__ b3,
    float* __restrict__ out_pad,      // [rows, 112]
    float* __restrict__ out_raw)      // [rows, 100] or nullptr
{
    __shared__ float Xl[16 * 516];
    __shared__ float H1[16 * 116];
    __shared__ float H2[16 * 116];

    const int tid  = threadIdx.x;
    const int wave = tid >> 5;
    const int lane = tid & 31;
    const int m    = lane & 15;
    const int hi   = lane >> 4;
    const int n    = wave * 16 + m;                 // 0..111
    const int nc   = (n < H_N) ? n : (H_N - 1);     // clamped, always valid
    const float nmask = (n < H_N) ? 1.0f : 0.0f;
    const long row0 = (long)blockIdx.x * 16;

    for (int idx = tid; idx < 16 * 512; idx += NTHREADS) {
        int r = idx >> 9, c = idx & 511;
        Xl[r * 516 + c] = X[(row0 + r) * 512 + c];
    }
    __syncthreads();

    // ---- layer 1: [16x512] @ [512x100->112] ----
    const float* w1p = W1 + nc;
    v8f acc = {0.f,0.f,0.f,0.f,0.f,0.f,0.f,0.f};
#pragma unroll 4
    for (int k0 = 0; k0 < F_N; k0 += 4) {
        const int kk = k0 + 2 * hi;
        v2f a; a.x = Xl[m * 516 + kk]; a.y = Xl[m * 516 + kk + 1];
        v2f b;
        b.x = nmask * w1p[kk * H_N];
        b.y = nmask * w1p[(kk + 1) * H_N];
        acc = __builtin_amdgcn_wmma_f32_16x16x4_f32(false, a, false, b,
                                                    (short)0, acc, false, false);
    }
    {
        const float bias = nmask * b1[nc];
#pragma unroll
        for (int r = 0; r < 8; ++r)
            H1[(r + 8 * hi) * 116 + n] = tanhf(acc[r] + bias);
    }
    __syncthreads();

    // ---- layer 2 ----
    const float* w2p = W2 + nc;
    v8f acc2 = {0.f,0.f,0.f,0.f,0.f,0.f,0.f,0.f};
#pragma unroll
    for (int k0 = 0; k0 < H_N; k0 += 4) {           // 25 iters, taps <= 99
        const int kk = k0 + 2 * hi;
        v2f a; a.x = H1[m * 116 + kk]; a.y = H1[m * 116 + kk + 1];
        v2f b;
        b.x = nmask * w2p[kk * H_N];
        b.y = nmask * w2p[(kk + 1) * H_N];
        acc2 = __builtin_amdgcn_wmma_f32_16x16x4_f32(false, a, false, b,
                                                     (short)0, acc2, false, false);
    }
    {
        const float bias = nmask * b2[nc];
#pragma unroll
        for (int r = 0; r < 8; ++r)
            H2[(r + 8 * hi) * 116 + n] = tanhf(acc2[r] + bias);
    }
    __syncthreads();

    // ---- layer 3 ----
    const float* w3p = W3 + nc;
    v8f acc3 = {0.f,0.f,0.f,0.f,0.f,0.f,0.f,0.f};
#pragma unroll
    for (int k0 = 0; k0 < H_N; k0 += 4) {
        const int kk = k0 + 2 * hi;
        v2f a; a.x = H2[m * 116 + kk]; a.y = H2[m * 116 + kk + 1];
        v2f b;
        b.x = nmask * w3p[kk * H_N];
        b.y = nmask * w3p[(kk + 1) * H_N];
        acc3 = __builtin_amdgcn_wmma_f32_16x16x4_f32(false, a, false, b,
                                                     (short)0, acc3, false, false);
    }
    {
        const float bias = nmask * b3[nc];
#pragma unroll
        for (int r = 0; r < 8; ++r) {
            const float v = tanhf(acc3[r] + bias);
            const long gr = row0 + r + 8 * hi;
            out_pad[gr * H_PAD + n] = v;            // cols 100..111 are 0
            if (out_raw != nullptr && n < H_N) out_raw[gr * H_N + n] = v;
        }
    }
}

// ---------------------------------------------------------------------------
// Z pass: XX-tile = intex^T @ f_irep via WMMA (kept in registers),
// Z = exp(rowdot(XX, f_urep)); fused partial sumA while the intex column
// block is still L2-hot. intex tile staged with async global->LDS b128.
// ---------------------------------------------------------------------------
__global__ __launch_bounds__(NTHREADS) void z_kernel(
    const float* __restrict__ intex,   // [B, E]
    const float* __restrict__ firep,   // [B, 112]
    const float* __restrict__ urep,    // [E, 112]
    float* __restrict__ Zg,            // [E]
    float* __restrict__ sumA)          // [B], pre-zeroed
{
    __shared__ __align__(16) float Al[64 * 20];   // row stride 80 B (16B mult)
    __shared__ float zbuf[16];
    __shared__ float zs[16];

    const int tid  = threadIdx.x;
    const int wave = tid >> 5;
    const int lane = tid & 31;
    const int m    = lane & 15;
    const int hi   = lane >> 4;
    const int n    = wave * 16 + m;
    const long e0  = (long)blockIdx.x * 16;

    if (tid < 16) zbuf[tid] = 0.0f;

    v8f acc = {0.f,0.f,0.f,0.f,0.f,0.f,0.f,0.f};
    for (int kc = 0; kc < B_N; kc += 64) {
        __syncthreads();
        // 64 rows x 16 floats, 4 x b128 per row = 256 async copies
        for (int idx = tid; idx < 256; idx += NTHREADS) {
            int k = idx >> 2, c4 = (idx & 3) << 2;
            async_copy_b128((unsigned)(size_t)&Al[k * 20 + c4],
                            &intex[(long)(kc + k) * E_N + e0 + c4]);
            if (kc + 64 < B_N)
                __builtin_prefetch(&intex[(long)(kc + 64 + k) * E_N + e0 + c4], 0, 1);
        }
        async_wait0();
        __syncthreads();
#pragma unroll
        for (int k0 = 0; k0 < 64; k0 += 4) {
            const int kk = k0 + 2 * hi;
            v2f a; a.x = Al[kk * 20 + m]; a.y = Al[(kk + 1) * 20 + m];
            v2f b;
            b.x = firep[(kc + kk) * H_PAD + n];
            b.y = firep[(kc + kk + 1) * H_PAD + n];
            acc = __builtin_amdgcn_wmma_f32_16x16x4_f32(false, a, false, b,
                                                        (short)0, acc, false, false);
        }
    }

#pragma unroll
    for (int r = 0; r < 8; ++r) {
        const long e = e0 + r + 8 * hi;
        atomicAdd(&zbuf[r + 8 * hi], acc[r] * urep[e * H_PAD + n]);
    }
    __syncthreads();
    if (tid < 16) {
        const float z = expf(zbuf[tid]);
        Zg[e0 + tid] = z;
        zs[tid] = z;
    }
    __syncthreads();

    // fused partial sumA[b] += sum_j intex[b, e0+j] * Z[e0+j]  (float4 reads)
    const float4* z4 = (const float4*)zs;
    float4 zv0 = z4[0], zv1 = z4[1], zv2 = z4[2], zv3 = z4[3];
    for (int b = tid; b < B_N; b += NTHREADS) {
        const float4* r4 = (const float4*)&intex[(long)b * E_N + e0];
        float4 a0 = r4[0], a1 = r4[1], a2 = r4[2], a3 = r4[3];
        float s = a0.x*zv0.x + a0.y*zv0.y + a0.z*zv0.z + a0.w*zv0.w
                + a1.x*zv1.x + a1.y*zv1.y + a1.z*zv1.z + a1.w*zv1.w
                + a2.x*zv2.x + a2.y*zv2.y + a2.z*zv2.z + a2.w*zv2.w
                + a3.x*zv3.x + a3.y*zv3.y + a3.z*zv3.z + a3.w*zv3.w;
        atomicAdd(&sumA[b], s);
    }
}

// ---------------------------------------------------------------------------
// sumB[e] = sum_b intex[b,e]*sumA[b];  normw[e] = Z[e]/sumB[e]
// ---------------------------------------------------------------------------
__global__ __launch_bounds__(256) void norm_kernel(
    const float* __restrict__ intex,
    const float* __restrict__ sumA,
    const float* __restrict__ Zg,
    float* __restrict__ normw)
{
    __shared__ float sA[B_N];
    const int tid = threadIdx.x;
    for (int i = tid; i < B_N; i += 256) sA[i] = sumA[i];
    __syncthreads();
    const long e = (long)blockIdx.x * 256 + tid;
    float s = 0.0f;
#pragma unroll 4
    for (int b = 0; b < B_N; ++b) s += intex[(long)b * E_N + e] * sA[b];
    normw[e] = Zg[e] / s;
}

// ---------------------------------------------------------------------------
// f_urep_out = intex @ (normw[e] * f_urep[e,:]) with 16-way K-split.
// intex tile staged with async global->LDS b128.
// mode 0: per-split partials (deterministic); mode 1: atomicAdd fallback.
// ---------------------------------------------------------------------------
__global__ __launch_bounds__(NTHREADS) void out_kernel(
    const float* __restrict__ intex,
    const float* __restrict__ urep,
    const float* __restrict__ normw,
    float* __restrict__ dst,
    int mode)
{
    __shared__ __align__(16) float Al[16 * 68];   // row stride 272 B (16B mult)

    const int tid  = threadIdx.x;
    const int wave = tid >> 5;
    const int lane = tid & 31;
    const int m    = lane & 15;
    const int hi   = lane >> 4;
    const int n    = wave * 16 + m;
    const long b0  = (long)blockIdx.x * 16;
    const int split = blockIdx.y;
    const long kbeg = (long)split * (E_N / KSPLIT);
    const long kend = kbeg + (E_N / KSPLIT);

    v8f acc = {0.f,0.f,0.f,0.f,0.f,0.f,0.f,0.f};
    for (long kc = kbeg; kc < kend; kc += 64) {
        __syncthreads();
        // 16 rows x 64 floats, 16 x b128 per row = 256 async copies
        for (int idx = tid; idx < 256; idx += NTHREADS) {
            int r = idx >> 4, k4 = (idx & 15) << 2;
            async_copy_b128((unsigned)(size_t)&Al[r * 68 + k4],
                            &intex[(b0 + r) * (long)E_N + kc + k4]);
            if (kc + 64 < kend)
                __builtin_prefetch(&intex[(b0 + r) * (long)E_N + kc + 64 + k4], 0, 1);
        }
        async_wait0();
        __syncthreads();
#pragma unroll
        for (int k0 = 0; k0 < 64; k0 += 4) {
            const int kk = k0 + 2 * hi;
            v2f a; a.x = Al[m * 68 + kk]; a.y = Al[m * 68 + kk + 1];
            const long kg = kc + kk;
            v2f b;
            b.x = normw[kg]     * urep[kg * H_PAD + n];
            b.y = normw[kg + 1] * urep[(kg + 1) * H_PAD + n];
            acc = __builtin_amdgcn_wmma_f32_16x16x4_f32(false, a, false, b,
                                                        (short)0, acc, false, false);
        }
    }
#pragma unroll
    for (int r = 0; r < 8; ++r) {
        const long row = b0 + r + 8 * hi;
        if (n < H_N) {
            if (mode == 0)
                dst[((size_t)split * B_N + row) * H_N + n] = acc[r];
            else
                atomicAdd(&dst[row * H_N + n], acc[r]);
        }
    }
}

// ---------------------------------------------------------------------------
__global__ __launch_bounds__(256) void reduce_kernel(
    const float* __restrict__ part, float* __restrict__ out, int nelem)
{
    int i = blockIdx.x * 256 + threadIdx.x;
    if (i >= nelem) return;
    float s = 0.0f;
#pragma unroll
    for (int k = 0; k < KSPLIT; ++k) s += part[(size_t)k * nelem + i];
    out[i] = s;
}

// ---------------------------------------------------------------------------
extern "C" void kernel_launch(void* const* d_in, const int* in_sizes, int n_in,
                              void* d_out, int out_size, void* d_ws, size_t ws_size,
                              hipStream_t stream) {
    const float* u_x   = (const float*)d_in[0];
    const float* i_x   = (const float*)d_in[1];
    const float* intex = (const float*)d_in[2];
    const float* W1    = (const float*)d_in[3];
    const float* b1    = (const float*)d_in[4];
    const float* W2    = (const float*)d_in[5];
    const float* b2    = (const float*)d_in[6];
    const float* W3    = (const float*)d_in[7];
    const float* b3    = (const float*)d_in[8];
    float* out = (float*)d_out;

    float* ws    = (float*)d_ws;
    float* urep  = ws;                                   // E * 112
    float* firep = urep  + (size_t)E_N * H_PAD;          // B * 112
    float* Zg    = firep + (size_t)B_N * H_PAD;          // E
    float* sumA  = Zg    + (size_t)E_N;                  // B
    float* normw = sumA  + (size_t)B_N;                  // E
    float* part  = normw + (size_t)E_N;                  // KSPLIT * B * 100

    const size_t base_f  = (size_t)E_N * H_PAD + (size_t)B_N * H_PAD
                         + 2 * (size_t)E_N + (size_t)B_N;
    const size_t part_f  = (size_t)KSPLIT * B_N * H_N;
    const bool have_part = ws_size >= (base_f + part_f) * sizeof(float);

    zero_kernel<<<(B_N + 255) / 256, 256, 0, stream>>>(sumA, B_N);

    mlp_kernel<<<E_N / 16, NTHREADS, 0, stream>>>(u_x, W1, b1, W2, b2, W3, b3,
                                                  urep, nullptr);
    mlp_kernel<<<B_N / 16, NTHREADS, 0, stream>>>(i_x, W1, b1, W2, b2, W3, b3,
                                                  firep, out + (size_t)B_N * H_N);

    z_kernel<<<E_N / 16, NTHREADS, 0, stream>>>(intex, firep, urep, Zg, sumA);

    norm_kernel<<<E_N / 256, 256, 0, stream>>>(intex, sumA, Zg, normw);

    dim3 og(B_N / 16, KSPLIT);
    if (have_part) {
        out_kernel<<<og, NTHREADS, 0, stream>>>(intex, urep, normw, part, 0);
        reduce_kernel<<<(B_N * H_N + 255) / 256, 256, 0, stream>>>(
            part, out, B_N * H_N);
    } else {
        zero_kernel<<<(B_N * H_N + 255) / 256, 256, 0, stream>>>(out, B_N * H_N);
        out_kernel<<<og, NTHREADS, 0, stream>>>(intex, urep, normw, out, 1);
    }
}